// VQVAEQuantize_37873021616666
// MI455X (gfx1250) — compile-verified
//
#include <hip/hip_runtime.h>
#include <hip/hip_bf16.h>

typedef __attribute__((ext_vector_type(2))) float v2f;
typedef __attribute__((ext_vector_type(8))) float v8f;
typedef __attribute__((ext_vector_type(4))) int   v4i;

#define NPIX 16384   // B*H*W = 16*32*32
#define CDIM 256     // input channels
#define EDIM 256     // embedding dim
#define NEMB 8192    // codebook size
#define HW   1024    // H*W
#define NCHUNK (NEMB / 64)
#define LPAD 260     // LDS row stride (floats): 16B-aligned rows, banks 4r mod 64 distinct
#define LOSS_SCALE 12.5f  // KLD_SCALE * (1 + COMMITMENT_COST)

#if __has_builtin(__builtin_amdgcn_global_load_async_to_lds_b128) && \
    __has_builtin(__builtin_amdgcn_s_wait_asynccnt)
#define HAS_ASYNC_LDS 1
typedef __attribute__((address_space(1))) v4i* gptr_v4i;   // global (AS1) int32x4*
typedef __attribute__((address_space(3))) v4i* sptr_v4i;   // LDS (AS3) int32x4*
#else
#define HAS_ASYNC_LDS 0
#endif

// ---------------------------------------------------------------------------
// K0: squared norms of codebook rows. One wave per code.
// ---------------------------------------------------------------------------
__global__ void k_norms(const float* __restrict__ embed, float* __restrict__ norms) {
    int lane = threadIdx.x & 31;
    int wave = threadIdx.x >> 5;
    int code = blockIdx.x * 8 + wave;
    const float* row = embed + (size_t)code * EDIM;
    float s = 0.f;
#pragma unroll
    for (int j = 0; j < 8; ++j) { float v = row[lane + j * 32]; s += v * v; }
#pragma unroll
    for (int m = 16; m >= 1; m >>= 1) s += __shfl_xor(s, m, 32);
    if (lane == 0) norms[code] = s;
}

// ---------------------------------------------------------------------------
// K1: projection GEMM  z_e[n,e] = sum_c z[n,c]*proj_w[e,c] + proj_b[e]
// One wave = one 16x16 WMMA tile, K=256 via 64x v_wmma_f32_16x16x4_f32.
// ---------------------------------------------------------------------------
__global__ void k_proj(const float* __restrict__ z, const float* __restrict__ pw,
                       const float* __restrict__ pb, float* __restrict__ ze) {
    __shared__ float a_lds[16 * LPAD];
    int tid  = threadIdx.x;
    int lane = tid & 31, wave = tid >> 5;
    int half = lane >> 4, l16 = lane & 15;
    int m_tile = blockIdx.x >> 1;                 // pixel tile (16 pixels)
    int n_tile = ((blockIdx.x & 1) << 3) + wave;  // embed-dim tile (16 dims)

    int pixbase = m_tile * 16;
    int b       = pixbase >> 10;       // batch index (HW=1024)
    int hwbase  = pixbase & 1023;
    const float* zb = z + (size_t)b * (CDIM * HW) + hwbase;

    // Stage A tile: a_lds[p][c] = z[b, c, hw(p)] ; coalesced (16 consecutive hw)
#pragma unroll
    for (int i = 0; i < 16; ++i) {
        int idx = tid + i * 256;
        int p = idx & 15, c = idx >> 4;
        a_lds[p * LPAD + c] = zb[(size_t)c * HW + p];
    }
    __syncthreads();

    const float* bw   = pw + (size_t)(n_tile * 16 + l16) * CDIM;  // proj_w row = B column
    const float* arow = &a_lds[l16 * LPAD];

    v8f acc = {0.f, 0.f, 0.f, 0.f, 0.f, 0.f, 0.f, 0.f};
#pragma unroll 8
    for (int kk = 0; kk < 64; ++kk) {
        int ko = kk * 4 + half * 2;              // lanes 0-15: K={0,1}; 16-31: K={2,3}
        v2f a  = *(const v2f*)(arow + ko);       // ds_read_b64
        v2f bm = *(const v2f*)(bw + ko);         // global_load_b64 (L2-hot)
        acc = __builtin_amdgcn_wmma_f32_16x16x4_f32(
            false, a, false, bm, (short)0, acc, false, false);
    }

    int   ncol = n_tile * 16 + l16;
    float bias = pb[ncol];
#pragma unroll
    for (int i = 0; i < 8; ++i) {                // C layout: VGPR i -> M = i + 8*half
        int m = m_tile * 16 + i + half * 8;
        ze[(size_t)m * EDIM + ncol] = acc[i] + bias;
    }
}

// ---------------------------------------------------------------------------
// Chunk copy: 64 codebook rows (64 KB) global -> LDS, b128 granularity.
// Async path uses GLOBAL_LOAD_ASYNC_TO_LDS_B128 (ASYNCcnt); fallback is
// synchronous b128 load/store.
// ---------------------------------------------------------------------------
static __device__ __forceinline__ void copy_chunk(const float* __restrict__ src,
                                                  float* dst, int tid) {
#pragma unroll
    for (int i = 0; i < 16; ++i) {
        int idx = tid + i * 256;       // 4096 x 16B per chunk
        int r   = idx >> 6;            // 64 b128 per 256-float row
        int c4  = (idx & 63) * 4;      // float offset within row (16B aligned)
#if HAS_ASYNC_LDS
        __builtin_amdgcn_global_load_async_to_lds_b128(
            (gptr_v4i)(src + (size_t)r * EDIM + c4),
            (sptr_v4i)(dst + r * LPAD + c4),
            0, 0);
#else
        float4 v = *(const float4*)(src + (size_t)r * EDIM + c4);
        *(float4*)(dst + r * LPAD + c4) = v;
#endif
    }
}

static __device__ __forceinline__ void wait_copy_barrier() {
#if HAS_ASYNC_LDS
    __builtin_amdgcn_s_wait_asynccnt(0);   // my wave's async copies landed
#endif
    __syncthreads();                       // => every wave's copies landed
}

// ---------------------------------------------------------------------------
// K2: fused distance GEMM + argmax.
//   score(n,code) = f.w - 0.5*||w||^2  (argmax score == argmin distance)
// Block: 64 pixels x all 8192 codes in 64-code chunks, double-buffered B.
// Each wave: one 16-pixel row tile x two 16-code column tiles, shared A frag,
// two independent WMMA accumulator chains.
// ---------------------------------------------------------------------------
__global__ void k_argmax(const float* __restrict__ ze, const float* __restrict__ embed,
                         const float* __restrict__ norms, int* __restrict__ ind_ws,
                         float* __restrict__ ind_out) {
    extern __shared__ float smem[];
    float* a_lds = smem;                      // 64*LPAD
    float* b_buf0 = smem + 64 * LPAD;         // 64*LPAD
    float* b_buf1 = b_buf0 + 64 * LPAD;       // 64*LPAD

    int tid  = threadIdx.x;
    int lane = tid & 31, wave = tid >> 5;
    int half = lane >> 4, l16 = lane & 15;
    int mt = wave >> 1;   // 0..3 : which 16-pixel row tile this wave owns
    int nh = wave & 1;    // 0..1 : which pair of 16-code column tiles

    int pixbase = blockIdx.x * 64;

    // Stage A: z_e tile [64 pixels x 256 dims], coalesced
#pragma unroll 4
    for (int i = 0; i < 64; ++i) {
        int idx = tid + i * 256;
        int r = idx >> 8, c = idx & 255;
        a_lds[r * LPAD + c] = ze[(size_t)(pixbase + r) * EDIM + c];
    }
    copy_chunk(embed, b_buf0, tid);           // chunk 0 in flight

    float best_v[8];
    int   best_i[8];
#pragma unroll
    for (int i = 0; i < 8; ++i) { best_v[i] = -3.4e38f; best_i[i] = 0; }

    const float* arow = &a_lds[(mt * 16 + l16) * LPAD];

    for (int chunk = 0; chunk < NCHUNK; ++chunk) {
        wait_copy_barrier();                  // buf[chunk&1] ready, prev compute done
        float* bcur = (chunk & 1) ? b_buf1 : b_buf0;
        if (chunk + 1 < NCHUNK)               // overlap next chunk's copy with compute
            copy_chunk(embed + (size_t)(chunk + 1) * 64 * EDIM,
                       ((chunk + 1) & 1) ? b_buf1 : b_buf0, tid);
        if (chunk + 2 < NCHUNK)               // global_prefetch_b8 hint for chunk+2
            __builtin_prefetch(embed + (size_t)(chunk + 2) * 64 * EDIM + tid * 64, 0, 0);

        const float* brow0 = &bcur[(nh * 32 + l16) * LPAD];  // column tile nh*2+0
        const float* brow1 = brow0 + 16 * LPAD;              // column tile nh*2+1
        v8f c0 = {0.f, 0.f, 0.f, 0.f, 0.f, 0.f, 0.f, 0.f};
        v8f c1 = {0.f, 0.f, 0.f, 0.f, 0.f, 0.f, 0.f, 0.f};
#pragma unroll 8
        for (int kk = 0; kk < 64; ++kk) {
            int ko = kk * 4 + half * 2;       // lanes 0-15: K={0,1}; 16-31: K={2,3}
            v2f a  = *(const v2f*)(arow + ko);    // shared A fragment
            v2f b0 = *(const v2f*)(brow0 + ko);
            v2f b1 = *(const v2f*)(brow1 + ko);
            c0 = __builtin_amdgcn_wmma_f32_16x16x4_f32(
                false, a, false, b0, (short)0, c0, false, false);
            c1 = __builtin_amdgcn_wmma_f32_16x16x4_f32(
                false, a, false, b1, (short)0, c1, false, false);
        }
        int   code0 = chunk * 64 + nh * 32 + l16;   // this lane's columns
        int   code1 = code0 + 16;
        float adj0 = 0.5f * norms[code0];
        float adj1 = 0.5f * norms[code1];
#pragma unroll
        for (int i = 0; i < 8; ++i) {         // ascending code order => first-hit ties
            float s0 = c0[i] - adj0;
            if (s0 > best_v[i]) { best_v[i] = s0; best_i[i] = code0; }
            float s1 = c1[i] - adj1;
            if (s1 > best_v[i]) { best_v[i] = s1; best_i[i] = code1; }
        }
    }

    // Reduce across the 16 lanes of each half (rows live per-half)
#pragma unroll
    for (int m = 8; m >= 1; m >>= 1) {
#pragma unroll
        for (int i = 0; i < 8; ++i) {
            float ov = __shfl_xor(best_v[i], m, 32);
            int   oi = __shfl_xor(best_i[i], m, 32);
            if (ov > best_v[i] || (ov == best_v[i] && oi < best_i[i])) {
                best_v[i] = ov; best_i[i] = oi;
            }
        }
    }

    __syncthreads();  // done reading a_lds; reuse smem[0..255] as combine buffer
    if (l16 == 0) {
#pragma unroll
        for (int i = 0; i < 8; ++i) {
            int mloc = mt * 16 + i + half * 8;
            smem[mloc * 4 + nh * 2 + 0] = best_v[i];
            smem[mloc * 4 + nh * 2 + 1] = __int_as_float(best_i[i]);
        }
    }
    __syncthreads();
    if (tid < 64) {   // combine the two waves sharing each pixel-row tile
        float v0 = smem[tid * 4 + 0]; int i0 = __float_as_int(smem[tid * 4 + 1]);
        float v1 = smem[tid * 4 + 2]; int i1 = __float_as_int(smem[tid * 4 + 3]);
        int bi = (v1 > v0 || (v1 == v0 && i1 < i0)) ? i1 : i0;
        int pix = pixbase + tid;
        ind_ws[pix]  = bi;
        ind_out[pix] = (float)bi;
    }
}

// ---------------------------------------------------------------------------
// K3: gather z_q = embed[ind], write NCHW output, per-pixel loss partial
// ---------------------------------------------------------------------------
__global__ void k_gather(const float* __restrict__ embed, const float* __restrict__ ze,
                         const int* __restrict__ ind, float* __restrict__ out,
                         float* __restrict__ part) {
    __shared__ float red[256];
    int pix = blockIdx.x, e = threadIdx.x;
    int code = ind[pix];
    float v = embed[(size_t)code * EDIM + e];
    float d = v - ze[(size_t)pix * EDIM + e];
    int b = pix >> 10, hw = pix & 1023;
    out[(size_t)b * (EDIM * HW) + (size_t)e * HW + hw] = v;  // [B,E,H,W]
    red[e] = d * d;
    __syncthreads();
#pragma unroll
    for (int s = 128; s > 0; s >>= 1) {
        if (e < s) red[e] += red[e + s];
        __syncthreads();
    }
    if (e == 0) part[pix] = red[0];
}

// ---------------------------------------------------------------------------
// K4: final loss reduction (deterministic, single block, no atomics)
// ---------------------------------------------------------------------------
__global__ void k_loss(const float* __restrict__ part, float* __restrict__ loss_out) {
    __shared__ float red[256];
    float s = 0.f;
    for (int i = threadIdx.x; i < NPIX; i += 256) s += part[i];
    red[threadIdx.x] = s;
    __syncthreads();
#pragma unroll
    for (int t = 128; t > 0; t >>= 1) {
        if (threadIdx.x < t) red[threadIdx.x] += red[threadIdx.x + t];
        __syncthreads();
    }
    if (threadIdx.x == 0)
        loss_out[0] = LOSS_SCALE * red[0] / (float)(NPIX * EDIM);
}

// ---------------------------------------------------------------------------
extern "C" void kernel_launch(void* const* d_in, const int* in_sizes, int n_in,
                              void* d_out, int out_size, void* d_ws, size_t ws_size,
                              hipStream_t stream) {
    const float* z  = (const float*)d_in[0];   // [16,256,32,32]
    const float* pw = (const float*)d_in[1];   // [256,256]
    const float* pb = (const float*)d_in[2];   // [256]
    const float* ew = (const float*)d_in[3];   // [8192,256]

    float* out      = (float*)d_out;
    float* zq_out   = out;                               // 4,194,304 floats
    float* loss_out = out + (size_t)NPIX * EDIM;         // 1 float
    float* ind_out  = loss_out + 1;                      // 16,384 floats

    float* ws    = (float*)d_ws;
    float* ze    = ws;                                   // 16 MB
    float* norms = ws + (size_t)NPIX * EDIM;             // 32 KB
    int*   indw  = (int*)(norms + NEMB);                 // 64 KB
    float* part  = (float*)(indw + NPIX);                // 64 KB

    // K0: code norms
    k_norms<<<NEMB / 8, 256, 0, stream>>>(ew, norms);
    // K1: projection GEMM (16384 tiles / 8 waves per block)
    k_proj<<<(NPIX / 16) * (EDIM / 16) / 8, 256, 0, stream>>>(z, pw, pb, ze);
    // K2: fused distance GEMM + argmax; dynamic LDS = A + double-buffered B
    size_t shbytes = (size_t)3 * 64 * LPAD * sizeof(float);
    k_argmax<<<NPIX / 64, 256, shbytes, stream>>>(ze, ew, norms, indw, ind_out);
    // K3: gather + NCHW store + loss partials
    k_gather<<<NPIX, 256, 0, stream>>>(ew, ze, indw, zq_out, part);
    // K4: loss scalar
    k_loss<<<1, 256, 0, stream>>>(part, loss_out);
}